// GATLayer_65687229825998
// MI455X (gfx1250) — compile-verified
//
#include <hip/hip_runtime.h>
#include <hip/hip_bf16.h>
#include <math.h>

// GAT layer for MI455X (gfx1250, wave32).
//   K1: h = x@W via V_WMMA_F32_16X16X4_F32 (fp32 WMMA; GEMM is BW-bound, keep full precision),
//       fused s = h@a1, t = h@a2 via wave32 shuffle reduction.
//   K2: zero d_out (h' accumulator) and expsum.
//   K3: per-edge scatter: w = exp(lrelu(s[src]+t[dst])); atomic fp32 adds into h'[src], expsum[src].
//   K4: finalize with self-loop term, divide.

typedef __attribute__((ext_vector_type(2))) float v2f;
typedef __attribute__((ext_vector_type(8))) float v8f;

#define IN_C  128
#define OUT_C 64
#define GAT_ALPHA 0.2f

__device__ __forceinline__ float lrelu(float z) {
    return z >= 0.0f ? z : GAT_ALPHA * z;
}

// ---------------------------------------------------------------------------
// Kernel 1: one wave per 16-row tile. D = 16x64 fp32 via 4 N-tiles x 32 K-steps
// of v_wmma_f32_16x16x4_f32. Fused s/t row-dots with shfl_xor reduction.
// ---------------------------------------------------------------------------
__global__ void gat_gemm_stw(const float* __restrict__ x,
                             const float* __restrict__ W,
                             const float* __restrict__ a,
                             float* __restrict__ h,
                             float* __restrict__ s,
                             float* __restrict__ t,
                             int n, int mtiles) {
    const int wavesPerBlock = blockDim.x >> 5;
    const int wave = blockIdx.x * wavesPerBlock + ((int)threadIdx.x >> 5);
    if (wave >= mtiles) return;              // wave-uniform: EXEC stays all-ones for WMMA

    const int lane = threadIdx.x & 31;
    const int half = lane >> 4;              // 0: K even pair / rows 0-7; 1: K odd pair / rows 8-15
    const int l15  = lane & 15;
    const int m0   = wave * 16;

    // A-fragment row for this lane (clamped for generic n; n=100000 is tile-exact)
    int arow = m0 + l15; if (arow > n - 1) arow = n - 1;
    const float* xrow = x + (size_t)arow * IN_C;

    // Preload all A fragments: lane holds A[m=l15][k = 4*ks + 2*half + j] in afrag[ks][j]
    v2f afrag[32];
#pragma unroll
    for (int ks = 0; ks < 32; ++ks) {
        const int k = ks * 4 + half * 2;
        afrag[ks].x = xrow[k];
        afrag[ks].y = xrow[k + 1];
    }

    float pr1[8], pr2[8];
#pragma unroll
    for (int r = 0; r < 8; ++r) { pr1[r] = 0.0f; pr2[r] = 0.0f; }

#pragma unroll
    for (int nt = 0; nt < 4; ++nt) {
        const int n0 = nt * 16;
        v8f acc = {};
#pragma unroll
        for (int ks = 0; ks < 32; ++ks) {
            const int k = ks * 4 + half * 2;
            // B[k][n]: row striped across lanes within a VGPR (mirror of A's K layout)
            v2f b;
            b.x = W[(size_t)k       * OUT_C + n0 + l15];
            b.y = W[(size_t)(k + 1) * OUT_C + n0 + l15];
            acc = __builtin_amdgcn_wmma_f32_16x16x4_f32(
                /*neg_a=*/false, afrag[ks], /*neg_b=*/false, b,
                /*c_mod=*/(short)0, acc, /*reuse_a=*/false, /*reuse_b=*/false);
        }
        // D layout: vgpr r, lane -> element [m = r + 8*half][n = n0 + l15]
        const float a1c = a[n0 + l15];
        const float a2c = a[OUT_C + n0 + l15];
#pragma unroll
        for (int r = 0; r < 8; ++r) {
            const int row = m0 + r + 8 * half;
            const float v = acc[r];
            if (row < n) h[(size_t)row * OUT_C + n0 + l15] = v;
            pr1[r] += v * a1c;
            pr2[r] += v * a2c;
        }
    }

    // Reduce over the 16 lanes of each half (xor masks < 16 never cross halves)
#pragma unroll
    for (int mask = 8; mask >= 1; mask >>= 1) {
#pragma unroll
        for (int r = 0; r < 8; ++r) {
            pr1[r] += __shfl_xor(pr1[r], mask, 32);
            pr2[r] += __shfl_xor(pr2[r], mask, 32);
        }
    }
    if (l15 == 0) {
#pragma unroll
        for (int r = 0; r < 8; ++r) {
            const int row = m0 + r + 8 * half;
            if (row < n) { s[row] = pr1[r]; t[row] = pr2[r]; }
        }
    }
}

// ---------------------------------------------------------------------------
// Kernel 2: zero the h' accumulator (d_out) and expsum.
// ---------------------------------------------------------------------------
__global__ void gat_zero(float* __restrict__ out, float* __restrict__ expsum, int n) {
    const size_t nout = (size_t)n * OUT_C;
    const size_t tot  = nout + (size_t)n;
    const size_t stride = (size_t)gridDim.x * blockDim.x;
    for (size_t i = (size_t)blockIdx.x * blockDim.x + threadIdx.x; i < tot; i += stride) {
        if (i < nout) out[i] = 0.0f;
        else          expsum[i - nout] = 0.0f;
    }
}

// ---------------------------------------------------------------------------
// Kernel 3: one wave per edge; lane handles 2 channels. fp32 atomics into L2.
// ---------------------------------------------------------------------------
__global__ void gat_edge(const int* __restrict__ ei,
                         const float* __restrict__ h,
                         const float* __restrict__ s,
                         const float* __restrict__ t,
                         float* __restrict__ hp,      // d_out used as accumulator
                         float* __restrict__ expsum,
                         int nE) {
    const int wavesPerBlock = blockDim.x >> 5;
    const int e = blockIdx.x * wavesPerBlock + ((int)threadIdx.x >> 5);
    if (e >= nE) return;                      // wave-uniform
    const int lane = threadIdx.x & 31;

    const int src = ei[e];
    const int dst = ei[nE + e];
    const float w = __expf(lrelu(s[src] + t[dst]));

    const float* hd = h  + (size_t)dst * OUT_C;
    float*       hs = hp + (size_t)src * OUT_C;
    const int c = lane * 2;
    atomicAdd(&hs[c],     w * hd[c]);
    atomicAdd(&hs[c + 1], w * hd[c + 1]);
    if (lane == 0) atomicAdd(&expsum[src], w);
}

// ---------------------------------------------------------------------------
// Kernel 4: add self-loop term (no atomics needed) and divide.
// ---------------------------------------------------------------------------
__global__ void gat_finalize(const float* __restrict__ h,
                             const float* __restrict__ s,
                             const float* __restrict__ t,
                             const float* __restrict__ expsum,
                             float* __restrict__ out,
                             int n) {
    const size_t tot = (size_t)n * OUT_C;
    const size_t stride = (size_t)gridDim.x * blockDim.x;
    for (size_t idx = (size_t)blockIdx.x * blockDim.x + threadIdx.x; idx < tot; idx += stride) {
        const size_t i = idx / OUT_C;
        const float w = __expf(lrelu(s[i] + t[i]));     // self-loop weight
        const float denom = expsum[i] + w;
        out[idx] = (out[idx] + w * h[idx]) / denom;
    }
}

extern "C" void kernel_launch(void* const* d_in, const int* in_sizes, int n_in,
                              void* d_out, int out_size, void* d_ws, size_t ws_size,
                              hipStream_t stream) {
    const float* x  = (const float*)d_in[0];   // (N, 128) fp32
    const float* W  = (const float*)d_in[1];   // (128, 64) fp32
    const float* a  = (const float*)d_in[2];   // (1, 128)  fp32
    const int*   ei = (const int*)d_in[3];     // (2, E)    int32 (harness convention)

    const int n = in_sizes[0] / IN_C;
    const int E = in_sizes[3] / 2;

    // Workspace layout (floats): h[N*64] | s[N] | t[N] | expsum[N]  (~26.8 MB)
    float* ws     = (float*)d_ws;
    float* h      = ws;
    float* s      = h + (size_t)n * OUT_C;
    float* t      = s + n;
    float* expsum = t + n;
    float* out    = (float*)d_out;             // doubles as h' accumulator

    const dim3 blk(256);                       // 8 wave32s per block
    const int wavesPerBlock = 256 / 32;

    const int mtiles = (n + 15) / 16;
    gat_gemm_stw<<<(mtiles + wavesPerBlock - 1) / wavesPerBlock, blk, 0, stream>>>(
        x, W, a, h, s, t, n, mtiles);

    gat_zero<<<1024, blk, 0, stream>>>(out, expsum, n);

    gat_edge<<<(E + wavesPerBlock - 1) / wavesPerBlock, blk, 0, stream>>>(
        ei, h, s, t, out, expsum, E);

    gat_finalize<<<2048, blk, 0, stream>>>(h, s, t, expsum, out, n);
}